// MD5Surrogate_27986006901475
// MI455X (gfx1250) — compile-verified
//
#include <hip/hip_runtime.h>

typedef __attribute__((ext_vector_type(2))) float v2f;
typedef __attribute__((ext_vector_type(8))) float v8f;

#define NR    64      // rounds
#define DH    256     // hidden dim
#define NB    16384   // batch rows
#define WAVES 8       // waves per block, each owns 16 rows
#define XS    25      // padded LDS stride for x tile (24 cols used)
#define HS    257     // padded LDS stride for hidden tile (256 cols)
#define XSF   (16 * XS)
#define HSF   (16 * HS)
#define WLDS  (XSF + 2 * HSF)   // floats of LDS per wave

// MD5 message schedule g(i) and shift amounts, precomputed
__constant__ int c_sched[64] = {
    0,1,2,3,4,5,6,7,8,9,10,11,12,13,14,15,
    1,6,11,0,5,10,15,4,9,14,3,8,13,2,7,12,
    5,8,11,14,1,4,7,10,13,0,3,6,9,12,15,2,
    0,7,14,5,12,3,10,1,8,15,6,13,4,11,2,9
};
__constant__ float c_shift[64] = {
    7,12,17,22, 7,12,17,22, 7,12,17,22, 7,12,17,22,
    5, 9,14,20, 5, 9,14,20, 5, 9,14,20, 5, 9,14,20,
    4,11,16,23, 4,11,16,23, 4,11,16,23, 4,11,16,23,
    6,10,15,21, 6,10,15,21, 6,10,15,21, 6,10,15,21
};

// Branchless exact-GELU: 0.5*x*(1+erf(x/sqrt(2))) with Abramowitz-Stegun
// 7.1.26 erf (|err| <= 1.5e-7). No branches -> no EXEC divergence near WMMA.
__device__ __forceinline__ float gelu_f(float x) {
    const float u  = x * 0.70710678118654752440f;
    const float ax = fabsf(u);
    const float t  = __builtin_amdgcn_rcpf(fmaf(0.3275911f, ax, 1.0f));
    float p = fmaf(t, 1.061405429f, -1.453152027f);
    p = fmaf(t, p, 1.421413741f);
    p = fmaf(t, p, -0.284496736f);
    p = fmaf(t, p, 0.254829592f);
    p = p * t;
    // exp(-ax^2) = 2^(-ax^2 * log2(e))  -> single v_exp_f32
    const float e = __builtin_amdgcn_exp2f(-ax * ax * 1.4426950408889634f);
    const float erf_ax = fmaf(-p, e, 1.0f);
    const float erfu   = copysignf(erf_ax, u);
    return 0.5f * x * (1.0f + erfu);
}

__global__ __launch_bounds__(WAVES * 32, 1)
void md5_mlp_wmma(const float* __restrict__ msg,
                  const float* __restrict__ st0,
                  const float* __restrict__ W1, const float* __restrict__ B1,
                  const float* __restrict__ W2, const float* __restrict__ B2,
                  const float* __restrict__ W3, const float* __restrict__ B3,
                  float* __restrict__ out)
{
    extern __shared__ float smem[];
    const int lane = threadIdx.x & 31;
    const int wave = threadIdx.x >> 5;
    const int half = lane >> 4;     // 0: lanes 0-15, 1: lanes 16-31
    const int l16  = lane & 15;

    float* xs = smem + wave * WLDS; // 16 x XS  (x tile, padded to K=24)
    float* hA = xs + XSF;           // 16 x HS  (layer-1 output)
    float* hB = hA + HSF;           // 16 x HS  (layer-2 output)

    const int mBase = (blockIdx.x * WAVES + wave) * 16;

    // Load initial state into xs cols 0..15 (C-layout: VGPR j -> rows j / j+8)
    #pragma unroll
    for (int j = 0; j < 8; ++j) {
        const int row = j + 8 * half;
        xs[row * XS + l16] = st0[(mBase + row) * 16 + l16];
    }
    // Zero-pad columns 22,23 once (never change)
    for (int idx = lane; idx < 32; idx += 32) {
        const int row = idx >> 1;
        xs[row * XS + 22 + (idx & 1)] = 0.0f;
    }

    #pragma unroll 1
    for (int r = 0; r < NR; ++r) {
        // ---- build x tile: cols 16..19 = scheduled word (uniform load, no branch)
        const int   g   = c_sched[r];
        const float ri0 = (float)r * (1.0f / 64.0f);
        const float ri1 = c_shift[r] * (1.0f / 25.0f);
        #pragma unroll
        for (int idx = lane; idx < 64; idx += 32) {     // 16 rows x 4 word cols
            const int row = idx >> 2;
            const int c   = idx & 3;
            xs[row * XS + 16 + c] = msg[(mBase + row) * 64 + g * 4 + c];
        }
        // cols 20,21 = rinfo (branchless select)
        for (int idx = lane; idx < 32; idx += 32) {     // 16 rows x 2 cols
            const int row = idx >> 1;
            xs[row * XS + 20 + (idx & 1)] = (idx & 1) ? ri1 : ri0;
        }

        // ---- layer 1: (16x24) @ W1(22x256); rows 22,23 zero via cndmask
        const float* __restrict__ w1 = W1 + (size_t)r * 22 * DH;
        const float* __restrict__ b1 = B1 + (size_t)r * DH;
        #pragma unroll 1
        for (int nt = 0; nt < 16; ++nt) {
            const int n = nt * 16 + l16;
            v8f acc = {};
            #pragma unroll
            for (int s = 0; s < 5; ++s) {               // K rows 0..19: no guard
                const int kk = 4 * s + 2 * half;
                v2f a; a.x = xs[l16 * XS + kk]; a.y = xs[l16 * XS + kk + 1];
                v2f b; b.x = w1[(kk) * DH + n]; b.y = w1[(kk + 1) * DH + n];
                acc = __builtin_amdgcn_wmma_f32_16x16x4_f32(
                          false, a, false, b, (short)0, acc, false, false);
            }
            {   // s = 5: lanes 0-15 use rows 20,21; lanes 16-31 contribute zero
                const int kk = 20 + 2 * half;           // 20 or 22
                v2f a; a.x = xs[l16 * XS + kk]; a.y = xs[l16 * XS + kk + 1];
                const float t0 = w1[20 * DH + n];       // always in bounds
                const float t1 = w1[21 * DH + n];
                v2f b; b.x = half ? 0.0f : t0; b.y = half ? 0.0f : t1;
                acc = __builtin_amdgcn_wmma_f32_16x16x4_f32(
                          false, a, false, b, (short)0, acc, false, false);
            }
            const float bias = b1[n];
            #pragma unroll
            for (int j = 0; j < 8; ++j)
                hA[(j + 8 * half) * HS + n] = gelu_f(acc[j] + bias);
        }

        // ---- layer 2: (16x256) @ W2(256x256); 4 N-tiles share each A load
        const float* __restrict__ w2 = W2 + (size_t)r * DH * DH;
        const float* __restrict__ b2 = B2 + (size_t)r * DH;
        #pragma unroll 1
        for (int ntg = 0; ntg < 4; ++ntg) {
            v8f acc0 = {}, acc1 = {}, acc2 = {}, acc3 = {};
            #pragma unroll 4
            for (int s = 0; s < 64; ++s) {
                const int kk = 4 * s + 2 * half;
                v2f a; a.x = hA[l16 * HS + kk]; a.y = hA[l16 * HS + kk + 1];
                #define L2STEP(u, accv) {                                      \
                    const int n = (ntg * 4 + (u)) * 16 + l16;                  \
                    v2f b;                                                     \
                    b.x = w2[(kk)     * DH + n];                               \
                    b.y = w2[(kk + 1) * DH + n];                               \
                    accv = __builtin_amdgcn_wmma_f32_16x16x4_f32(              \
                               false, a, false, b, (short)0, accv, false, false); }
                L2STEP(0, acc0) L2STEP(1, acc1) L2STEP(2, acc2) L2STEP(3, acc3)
                #undef L2STEP
            }
            #define L2OUT(u, accv) {                                           \
                const int n = (ntg * 4 + (u)) * 16 + l16;                      \
                const float bias = b2[n];                                      \
                for (int j = 0; j < 8; ++j)                                    \
                    hB[(j + 8 * half) * HS + n] = gelu_f(accv[j] + bias); }
            L2OUT(0, acc0) L2OUT(1, acc1) L2OUT(2, acc2) L2OUT(3, acc3)
            #undef L2OUT
        }

        // ---- layer 3: (16x256) @ W3(256x16), single N-tile
        const float* __restrict__ w3 = W3 + (size_t)r * DH * 16;
        const float* __restrict__ b3 = B3 + (size_t)r * 16;
        v8f acc = {};
        #pragma unroll 4
        for (int s = 0; s < 64; ++s) {
            const int kk = 4 * s + 2 * half;
            v2f a; a.x = hB[l16 * HS + kk]; a.y = hB[l16 * HS + kk + 1];
            v2f b; b.x = w3[(kk) * 16 + l16]; b.y = w3[(kk + 1) * 16 + l16];
            acc = __builtin_amdgcn_wmma_f32_16x16x4_f32(
                      false, a, false, b, (short)0, acc, false, false);
        }
        const float bias3 = b3[l16];
        if (r < NR - 1) {
            // next round's state back into xs cols 0..15
            #pragma unroll
            for (int j = 0; j < 8; ++j)
                xs[(j + 8 * half) * XS + l16] = acc[j] + bias3;
        } else {
            #pragma unroll
            for (int j = 0; j < 8; ++j)
                out[(mBase + j + 8 * half) * 16 + l16] = acc[j] + bias3;
        }
    }
}

extern "C" void kernel_launch(void* const* d_in, const int* in_sizes, int n_in,
                              void* d_out, int out_size, void* d_ws, size_t ws_size,
                              hipStream_t stream) {
    const float* msg = (const float*)d_in[0];
    const float* st0 = (const float*)d_in[1];
    const float* W1  = (const float*)d_in[2];
    const float* B1  = (const float*)d_in[3];
    const float* W2  = (const float*)d_in[4];
    const float* B2  = (const float*)d_in[5];
    const float* W3  = (const float*)d_in[6];
    const float* B3  = (const float*)d_in[7];
    float* out = (float*)d_out;

    dim3 grid(NB / (WAVES * 16));     // 128 blocks x 8 waves x 16 rows = 16384
    dim3 block(WAVES * 32);           // 256 threads = 8 waves (wave32)
    size_t shmem = (size_t)WLDS * WAVES * sizeof(float);  // ~270 KB dynamic LDS

    md5_mlp_wmma<<<grid, block, shmem, stream>>>(msg, st0, W1, B1, W2, B2, W3, B3, out);
}